// GCN_5566277616457
// MI455X (gfx1250) — compile-verified
//
#include <hip/hip_runtime.h>
#include <math.h>

typedef __attribute__((ext_vector_type(2))) float v2f;
typedef __attribute__((ext_vector_type(8))) float v8f;

#define NN 50000
#define H  64

// ---------------- utility kernels ----------------

__global__ void k_zero(float* __restrict__ p, int n) {
    int i = blockIdx.x * blockDim.x + threadIdx.x;
    if (i < n) p[i] = 0.0f;
}

__global__ void k_deg(const int* __restrict__ dst, float* __restrict__ deg, int E) {
    int e = blockIdx.x * blockDim.x + threadIdx.x;
    if (e < E) atomicAdd(&deg[dst[e]], 1.0f);
}

// dinv[i] = rsqrt(deg[i] + 1)   (+1 = self loop; always > 0)
__global__ void k_dinv(float* __restrict__ deg_dinv, int n) {
    int i = blockIdx.x * blockDim.x + threadIdx.x;
    if (i < n) deg_dinv[i] = rsqrtf(deg_dinv[i] + 1.0f);
}

__global__ void k_norm(const int* __restrict__ src, const int* __restrict__ dst,
                       const float* __restrict__ dinv, float* __restrict__ norm, int E) {
    int e = blockIdx.x * blockDim.x + threadIdx.x;
    if (e < E) norm[e] = dinv[src[e]] * dinv[dst[e]];
}

// ---------------- WMMA GEMM: C[M,NB] = A[M,K] @ B[K,NB] (+ bias) ----------------
// One wave32 per 16 x (NT*16) output block; M multiple of 16 (50000 = 3125*16).
// K, NB compile-time -> immediate-offset loads, no exec masking in the hot loop.
// A 16x4 f32 layout: lane L -> row (L&15); VGPR0 holds K=(L>>4)*2, VGPR1 K=(L>>4)*2+1.
// B 4x16 f32 mirrored. C/D: VGPR i, lanes0-15 -> M=i, lanes16-31 -> M=i+8.
template<int K, int NB, int NT>
__global__ void k_gemm_wmma(const float* __restrict__ A, const float* __restrict__ B,
                            const float* __restrict__ bias, float* __restrict__ C) {
    const int tm   = blockIdx.x;
    const int lane = threadIdx.x;
    const int half = lane >> 4;         // 0|1
    const int l16  = lane & 15;
    const int arow = tm * 16 + l16;

    const float* Ap = A + (size_t)arow * K + half * 2;  // + imm k*4 bytes
    const float* Bp = B + half * 2 * NB;                // + imm (k*NB + col)*4

    v8f acc[NT];
#pragma unroll
    for (int t = 0; t < NT; ++t) acc[t] = (v8f){};

    for (int k = 0; k < K; k += 4) {
        v2f a;
        a.x = Ap[k];
        a.y = Ap[k + 1];
#pragma unroll
        for (int t = 0; t < NT; ++t) {
            const int c = t * 16 + l16;
            v2f b;
            if constexpr (NT * 16 == NB) {
                b.x = Bp[k * NB + c];
                b.y = Bp[k * NB + NB + c];
            } else {
                // branchless tail masking (only the padded classifier tile)
                const float m  = (c < NB) ? 1.0f : 0.0f;
                const int   cc = (c < NB) ? c : (NB - 1);
                b.x = m * Bp[k * NB + cc];
                b.y = m * Bp[k * NB + NB + cc];
            }
            acc[t] = __builtin_amdgcn_wmma_f32_16x16x4_f32(
                false, a, false, b, (short)0, acc[t], false, false);
        }
    }

#pragma unroll
    for (int t = 0; t < NT; ++t) {
        const int c = t * 16 + l16;
        if (NT * 16 == NB || c < NB) {
            const float bv = bias ? bias[c] : 0.0f;
#pragma unroll
            for (int i = 0; i < 8; ++i)
                C[(size_t)(tm * 16 + i + half * 8) * NB + c] = acc[t][i] + bv;
        }
    }
}

// ---------------- aggregation ----------------

// agg[i,:] = dinv[i]^2 * h[i,:]   (self-loop term, also initializes agg)
__global__ void k_selfloop_init(const float* __restrict__ h, const float* __restrict__ dinv,
                                float* __restrict__ agg, int n) {
    int i = blockIdx.x * blockDim.x + threadIdx.x;
    if (i >= n) return;
    float di = dinv[i >> 6];            // F = 64
    agg[i] = di * di * h[i];
}

// one wave per edge: agg[dst,:] += norm[e] * h[src,:]  (64 floats, float2/lane)
__global__ void k_edge_scatter(const int* __restrict__ src, const int* __restrict__ dst,
                               const float* __restrict__ norm, const float* __restrict__ h,
                               float* __restrict__ agg, int E) {
    int gid  = blockIdx.x * blockDim.x + threadIdx.x;
    int e    = gid >> 5;
    int lane = gid & 31;
    if (e >= E) return;
    const int   s = src[e];
    const int   d = dst[e];
    const float w = norm[e];
    const float2 v = ((const float2*)(h + (size_t)s * H))[lane];
    float* o = agg + (size_t)d * H + lane * 2;
    atomicAdd(o,     w * v.x);
    atomicAdd(o + 1, w * v.y);
}

// act = (agg + bias) with optional ELU (alpha = 1)
__global__ void k_bias_act(const float* __restrict__ agg, const float* __restrict__ bias,
                           float* __restrict__ act, int n, int do_elu) {
    int i = blockIdx.x * blockDim.x + threadIdx.x;
    if (i >= n) return;
    float v = agg[i] + bias[i & (H - 1)];
    act[i] = (do_elu && v <= 0.0f) ? expm1f(v) : v;
}

// ---------------- driver ----------------

extern "C" void kernel_launch(void* const* d_in, const int* in_sizes, int n_in,
                              void* d_out, int out_size, void* d_ws, size_t ws_size,
                              hipStream_t stream) {
    const float* x  = (const float*)d_in[0];
    const int*   ei = (const int*)d_in[1];
    const float* W1 = (const float*)d_in[2];
    const float* b1 = (const float*)d_in[3];
    const float* W2 = (const float*)d_in[4];
    const float* b2 = (const float*)d_in[5];
    const float* W3 = (const float*)d_in[6];
    const float* b3 = (const float*)d_in[7];
    const float* Wl = (const float*)d_in[8];
    const float* bl = (const float*)d_in[9];
    float* out = (float*)d_out;

    const int E = in_sizes[1] / 2;       // 800000
    const int* src = ei;
    const int* dst = ei + E;

    // workspace layout (floats)
    float* dinv = (float*)d_ws;          // NN (deg, then dinv in place)
    float* norm = dinv + NN;             // E
    float* G    = norm + E;              // NN*H  (gemm output)
    float* AGG  = G + (size_t)NN * H;    // NN*H
    float* ACT  = AGG + (size_t)NN * H;  // NN*H

    const int NF = NN * H;               // 3.2M elems per feature buffer
    const dim3 blk(256);

    // --- normalization coefficients ---
    k_zero<<<dim3((NN + 255) / 256), blk, 0, stream>>>(dinv, NN);
    k_deg<<<dim3((E + 255) / 256), blk, 0, stream>>>(dst, dinv, E);
    k_dinv<<<dim3((NN + 255) / 256), blk, 0, stream>>>(dinv, NN);
    k_norm<<<dim3((E + 255) / 256), blk, 0, stream>>>(src, dst, dinv, norm, E);

    const dim3 gGemm(NN / 16);           // 3125 waves, each does 16 x 64
    const dim3 wave(32);
    const dim3 gElem((NF + 255) / 256);
    const dim3 gScat((E * 32 + 255) / 256);

    // --- layer 1: K=128 ---
    k_gemm_wmma<128, H, 4><<<gGemm, wave, 0, stream>>>(x, W1, nullptr, G);
    k_selfloop_init<<<gElem, blk, 0, stream>>>(G, dinv, AGG, NF);
    k_edge_scatter<<<gScat, blk, 0, stream>>>(src, dst, norm, G, AGG, E);
    k_bias_act<<<gElem, blk, 0, stream>>>(AGG, b1, ACT, NF, 1);

    // --- layer 2: K=64 ---
    k_gemm_wmma<H, H, 4><<<gGemm, wave, 0, stream>>>(ACT, W2, nullptr, G);
    k_selfloop_init<<<gElem, blk, 0, stream>>>(G, dinv, AGG, NF);
    k_edge_scatter<<<gScat, blk, 0, stream>>>(src, dst, norm, G, AGG, E);
    k_bias_act<<<gElem, blk, 0, stream>>>(AGG, b2, ACT, NF, 1);

    // --- layer 3: K=64, no ELU after ---
    k_gemm_wmma<H, H, 4><<<gGemm, wave, 0, stream>>>(ACT, W3, nullptr, G);
    k_selfloop_init<<<gElem, blk, 0, stream>>>(G, dinv, AGG, NF);
    k_edge_scatter<<<gScat, blk, 0, stream>>>(src, dst, norm, G, AGG, E);
    k_bias_act<<<gElem, blk, 0, stream>>>(AGG, b3, ACT, NF, 0);

    // --- classifier: [NN,64] @ [64,40] + bl, 3 N-tiles with tail masking ---
    k_gemm_wmma<H, 40, 3><<<gGemm, wave, 0, stream>>>(ACT, Wl, bl, out);
}